// ProduceEdges_6038724018272
// MI455X (gfx1250) — compile-verified
//
#include <hip/hip_runtime.h>
#include <stdint.h>

typedef __attribute__((ext_vector_type(2))) float v2f;
typedef __attribute__((ext_vector_type(8))) float v8f;

#define BATCH 8
#define SEQ   2048
#define DIM   256
#define NBUCK 4096          // 12-bit radix buckets per level
#define CAP   4096          // candidate cap per batch (need ~1024+ties)
#define TILEWAVES 1056      // upper-triangle 32x64 tiles per batch

// ---------------- zero scratch counters/histograms ----------------
__global__ void zero_kernel(unsigned* __restrict__ p, unsigned n) {
  unsigned i = blockIdx.x * blockDim.x + threadIdx.x;
  if (i < n) p[i] = 0u;
}

// ---------------- squared row norms: sq[b*SEQ + i] ----------------
__global__ void norms_kernel(const float* __restrict__ E, float* __restrict__ sq) {
  const int wave = (int)((blockIdx.x * blockDim.x + threadIdx.x) >> 5);
  const int lane = threadIdx.x & 31;
  const float4* r4 = (const float4*)(E + (size_t)wave * DIM);
  float4 x = r4[lane];
  float4 y = r4[lane + 32];
  float s = x.x*x.x + x.y*x.y + x.z*x.z + x.w*x.w
          + y.x*y.x + y.y*y.y + y.z*y.z + y.w*y.w;
  #pragma unroll
  for (int off = 16; off > 0; off >>= 1) s += __shfl_xor(s, off, 32);
  if (lane == 0) sq[wave] = s;
}

// ---- f32 WMMA GEMM: one wave computes a 32x64 tile of d2 (upper tri only) ----
// 2 A-frags x 4 B-frags = 8 WMMAs per K-step of 4; 6 loads per 8 WMMAs.
__global__ __launch_bounds__(256) void gemm_d2_kernel(const float* __restrict__ E,
                                                      const float* __restrict__ sq,
                                                      float* __restrict__ d2) {
  const int gwave = (int)((blockIdx.x * blockDim.x + threadIdx.x) >> 5);
  const int lane  = threadIdx.x & 31;
  int b = gwave / TILEWAVES;
  int r = gwave - b * TILEWAVES;
  // triangular decode: 32-row band ti covers 64-col tiles tc in [ti>>1, 32)
  int ti = 0, rowlen = 32;
  while (r >= rowlen) { r -= rowlen; ++ti; rowlen = 32 - (ti >> 1); }
  const int tc = (ti >> 1) + r;
  const int i0 = ti * 32, j0 = tc * 64;
  const int m = lane & 15, half = lane >> 4;

  const float* Eb = E + (size_t)b * SEQ * DIM;
  // A frag (16x4 f32): lane holds A[m][2*half+{0,1}]; B frag (4x16): B[2*half+r][m]=E[col][k]
  const float* pa[2];
  const float* pb[4];
  #pragma unroll
  for (int f = 0; f < 2; ++f) pa[f] = Eb + (size_t)(i0 + 16 * f + m) * DIM + 2 * half;
  #pragma unroll
  for (int f = 0; f < 4; ++f) pb[f] = Eb + (size_t)(j0 + 16 * f + m) * DIM + 2 * half;

  v8f acc[2][4];
  #pragma unroll
  for (int fi = 0; fi < 2; ++fi)
    #pragma unroll
    for (int fj = 0; fj < 4; ++fj) acc[fi][fj] = (v8f){};

  for (int kb = 0; kb < DIM; kb += 4) {
    v2f af[2], bf[4];
    #pragma unroll
    for (int f = 0; f < 2; ++f) af[f] = *(const v2f*)(pa[f] + kb);
    #pragma unroll
    for (int f = 0; f < 4; ++f) bf[f] = *(const v2f*)(pb[f] + kb);
    #pragma unroll
    for (int fi = 0; fi < 2; ++fi)
      #pragma unroll
      for (int fj = 0; fj < 4; ++fj)
        acc[fi][fj] = __builtin_amdgcn_wmma_f32_16x16x4_f32(
            false, af[fi], false, bf[fj], (short)0, acc[fi][fj], false, false);
  }

  const float* sqb = sq + b * SEQ;
  float* db = d2 + (size_t)b * SEQ * SEQ;
  #pragma unroll
  for (int fi = 0; fi < 2; ++fi) {
    #pragma unroll
    for (int fj = 0; fj < 4; ++fj) {
      const int jc = j0 + 16 * fj + m;
      const float sj = sqb[jc];
      #pragma unroll
      for (int v = 0; v < 8; ++v) {
        const int ir = i0 + 16 * fi + v + 8 * half;   // C/D layout: row = v + 8*(lane>=16)
        float dd = sqb[ir] + sj - 2.0f * acc[fi][fj][v];
        db[(size_t)ir * SEQ + jc] = fmaxf(dd, 0.0f);
      }
    }
  }
}

// ---------------- level-1 histogram on bits[30:19] (upper triangle) ----------------
__global__ void hist1_kernel(const float* __restrict__ d2, unsigned* __restrict__ hist) {
  __shared__ unsigned lh[NBUCK];
  const int b = blockIdx.y;
  for (int t = threadIdx.x; t < NBUCK; t += blockDim.x) lh[t] = 0u;
  __syncthreads();
  const float* db = d2 + (size_t)b * SEQ * SEQ;
  const unsigned base = blockIdx.x * blockDim.x + threadIdx.x;   // < 65536
  for (int it = 0; it < 64; ++it) {
    const unsigned idx = base + (unsigned)it * 65536u;
    const unsigned i = idx >> 11, j = idx & 2047u;
    if (j > i) {
      const unsigned bits = __float_as_uint(db[idx]) & 0x7fffffffu;
      atomicAdd(&lh[bits >> 19], 1u);
    }
  }
  __syncthreads();
  for (int t = threadIdx.x; t < NBUCK; t += blockDim.x) {
    const unsigned v = lh[t];
    if (v) atomicAdd(&hist[b * NBUCK + t], v);
  }
}

// ------- level-2 histogram on bits[18:7] for elements in level-1 bucket T1 -------
__global__ void hist2_kernel(const float* __restrict__ d2, const unsigned* __restrict__ T1,
                             unsigned* __restrict__ hist) {
  __shared__ unsigned lh[NBUCK];
  const int b = blockIdx.y;
  const unsigned t1 = T1[b];
  for (int t = threadIdx.x; t < NBUCK; t += blockDim.x) lh[t] = 0u;
  __syncthreads();
  const float* db = d2 + (size_t)b * SEQ * SEQ;
  const unsigned base = blockIdx.x * blockDim.x + threadIdx.x;
  for (int it = 0; it < 64; ++it) {
    const unsigned idx = base + (unsigned)it * 65536u;
    const unsigned i = idx >> 11, j = idx & 2047u;
    if (j > i) {
      const unsigned bits = __float_as_uint(db[idx]) & 0x7fffffffu;
      if ((bits >> 19) == t1) atomicAdd(&lh[(bits >> 7) & 0xfffu], 1u);
    }
  }
  __syncthreads();
  for (int t = threadIdx.x; t < NBUCK; t += blockDim.x) {
    const unsigned v = lh[t];
    if (v) atomicAdd(&hist[b * NBUCK + t], v);
  }
}

// ------- find smallest bucket T with cum(<=T) >= k; outCum = cum(<T) -------
__global__ void findT_kernel(const unsigned* __restrict__ hist, unsigned kBase,
                             const unsigned* __restrict__ prevCum, int usePrev,
                             unsigned* __restrict__ outT, unsigned* __restrict__ outCum) {
  const int b = blockIdx.x;
  const unsigned k = kBase - (usePrev ? prevCum[b] : 0u);
  __shared__ unsigned ssum[1024];
  const int t = threadIdx.x;               // 1024 threads, 4 buckets each
  unsigned h[4], tot = 0u;
  #pragma unroll
  for (int rr = 0; rr < 4; ++rr) { h[rr] = hist[b * NBUCK + 4 * t + rr]; tot += h[rr]; }
  ssum[t] = tot;
  __syncthreads();
  for (int off = 1; off < 1024; off <<= 1) {       // Hillis-Steele inclusive scan
    const unsigned v = ssum[t];
    const unsigned a = (t >= off) ? ssum[t - off] : 0u;
    __syncthreads();
    ssum[t] = v + a;
    __syncthreads();
  }
  const unsigned incl = ssum[t], excl = incl - tot;
  if (excl < k && incl >= k) {                     // unique thread
    unsigned c = excl;
    #pragma unroll
    for (int rr = 0; rr < 4; ++rr) {
      if (c + h[rr] >= k) { outT[b] = 4 * t + rr; outCum[b] = c; break; }
      c += h[rr];
    }
  }
}

// ------- collect candidates with 24-bit key <= T24 into per-batch buffer -------
__global__ void collect_kernel(const float* __restrict__ d2, const unsigned* __restrict__ T1,
                               const unsigned* __restrict__ T2, unsigned* __restrict__ counts,
                               unsigned long long* __restrict__ cand) {
  const int b = blockIdx.y;
  const unsigned T24 = (T1[b] << 12) | T2[b];
  const float* db = d2 + (size_t)b * SEQ * SEQ;
  const unsigned base = blockIdx.x * blockDim.x + threadIdx.x;
  for (int it = 0; it < 64; ++it) {
    const unsigned idx = base + (unsigned)it * 65536u;
    const unsigned i = idx >> 11, j = idx & 2047u;
    if (j > i) {
      const unsigned bits = __float_as_uint(db[idx]) & 0x7fffffffu;
      if ((bits >> 7) <= T24) {
        const unsigned pos = atomicAdd(&counts[b], 1u);
        if (pos < CAP)
          cand[(size_t)b * CAP + pos] = ((unsigned long long)bits << 32) | (unsigned long long)idx;
      }
    }
  }
}

// ------- per-batch bitonic sort of candidates, emit the kPairs closest pairs -------
__global__ void sortk_kernel(const unsigned* __restrict__ counts,
                             const unsigned long long* __restrict__ cand,
                             unsigned* __restrict__ pairs, int kPairs) {
  const int b = blockIdx.x;
  __shared__ unsigned long long key[CAP];
  unsigned n = counts[b]; if (n > CAP) n = CAP;
  for (int t = threadIdx.x; t < CAP; t += blockDim.x)
    key[t] = (t < (int)n) ? cand[(size_t)b * CAP + t] : 0xFFFFFFFFFFFFFFFFull;
  __syncthreads();
  for (int kk = 2; kk <= CAP; kk <<= 1) {
    for (int j = kk >> 1; j > 0; j >>= 1) {
      for (int idx = threadIdx.x; idx < CAP; idx += blockDim.x) {
        const int ixj = idx ^ j;
        if (ixj > idx) {
          const unsigned long long a = key[idx], c = key[ixj];
          const bool up = ((idx & kk) == 0);
          if (up ? (a > c) : (a < c)) { key[idx] = c; key[ixj] = a; }
        }
      }
      __syncthreads();
    }
  }
  for (int t = threadIdx.x; t < kPairs; t += blockDim.x) {
    const unsigned flat = (unsigned)(key[t] & 0xFFFFFFFFull);
    pairs[(b * kPairs + t) * 2 + 0] = flat >> 11;      // i (row, i<j)
    pairs[(b * kPairs + t) * 2 + 1] = flat & 2047u;    // j (col)
  }
}

// ------- emit output: zeros for diagonal picks, then (e_i-e_j, e_j-e_i) per pair -------
__global__ void emit_kernel(const float* __restrict__ E, const unsigned* __restrict__ pairs,
                            float* __restrict__ out, int nE, int kPairs) {
  const int eb = blockIdx.x;
  const int b  = eb / nE;
  const int e  = eb - b * nE;
  float* o = out + (size_t)eb * DIM + threadIdx.x;
  const int nz = nE - 2 * kPairs;          // = SEQ diagonal zeros
  if (e < nz) { *o = 0.0f; return; }
  const int t = (e - nz) >> 1;
  unsigned r = pairs[(b * kPairs + t) * 2 + 0];
  unsigned c = pairs[(b * kPairs + t) * 2 + 1];
  if ((e - nz) & 1) { unsigned tmp = r; r = c; c = tmp; }
  const float* Eb = E + (size_t)b * SEQ * DIM;
  *o = Eb[(size_t)r * DIM + threadIdx.x] - Eb[(size_t)c * DIM + threadIdx.x];
}

extern "C" void kernel_launch(void* const* d_in, const int* in_sizes, int n_in,
                              void* d_out, int out_size, void* d_ws, size_t ws_size,
                              hipStream_t stream) {
  const float* E = (const float*)d_in[0];
  float* out = (float*)d_out;
  char* ws = (char*)d_ws;

  const int nE = out_size / (BATCH * DIM);                     // 4096 edges per batch
  const int kPairs = (nE > SEQ) ? (nE - SEQ) / 2 : 0;          // 1024 closest pairs

  // workspace layout (~135 MB)
  size_t offD2   = 0;
  size_t offH1   = offD2  + (size_t)BATCH * SEQ * SEQ * 4;     // d2 matrix (upper tri valid)
  size_t offH2   = offH1  + (size_t)BATCH * NBUCK * 4;
  size_t offCnt  = offH2  + (size_t)BATCH * NBUCK * 4;
  size_t offT1   = offCnt + (size_t)BATCH * 4;
  size_t offC1   = offT1  + (size_t)BATCH * 4;
  size_t offT2   = offC1  + (size_t)BATCH * 4;
  size_t offC2   = offT2  + (size_t)BATCH * 4;
  size_t offCand = offC2  + (size_t)BATCH * 4;
  size_t offPair = offCand + (size_t)BATCH * CAP * 8;
  size_t offSq   = offPair + (size_t)BATCH * 4096 * 2 * 4;

  float*    d2p   = (float*)(ws + offD2);
  unsigned* h1p   = (unsigned*)(ws + offH1);
  unsigned* h2p   = (unsigned*)(ws + offH2);
  unsigned* cntp  = (unsigned*)(ws + offCnt);
  unsigned* t1p   = (unsigned*)(ws + offT1);
  unsigned* c1p   = (unsigned*)(ws + offC1);
  unsigned* t2p   = (unsigned*)(ws + offT2);
  unsigned* c2p   = (unsigned*)(ws + offC2);
  unsigned long long* candp = (unsigned long long*)(ws + offCand);
  unsigned* pairp = (unsigned*)(ws + offPair);
  float*    sqp   = (float*)(ws + offSq);

  // 1) zero hist1 + hist2 + counters (+T/C slots) every call (graph replays)
  const unsigned nZero = (unsigned)((offCand - offH1) / 4);
  zero_kernel<<<(nZero + 255) / 256, 256, 0, stream>>>((unsigned*)(ws + offH1), nZero);

  // 2) row squared norms (one wave32 per row, 8 rows per 256-thread block)
  norms_kernel<<<(BATCH * SEQ) / 8, 256, 0, stream>>>(E, sqp);

  // 3) WMMA f32 GEMM -> d2 (upper-triangle 32x64 tiles; 8 waves per block)
  gemm_d2_kernel<<<(BATCH * TILEWAVES) / 8, 256, 0, stream>>>(E, sqp, d2p);

  if (kPairs > 0) {
    dim3 scanGrid(256, BATCH);
    // 4) level-1 radix histogram + threshold
    hist1_kernel<<<scanGrid, 256, 0, stream>>>(d2p, h1p);
    findT_kernel<<<BATCH, 1024, 0, stream>>>(h1p, (unsigned)kPairs, c1p, 0, t1p, c1p);
    // 5) level-2 refinement within threshold bucket
    hist2_kernel<<<scanGrid, 256, 0, stream>>>(d2p, t1p, h2p);
    findT_kernel<<<BATCH, 1024, 0, stream>>>(h2p, (unsigned)kPairs, c1p, 1, t2p, c2p);
    // 6) gather candidates (~kPairs + ties), sort, extract pairs
    collect_kernel<<<scanGrid, 256, 0, stream>>>(d2p, t1p, t2p, cntp, candp);
    sortk_kernel<<<BATCH, 1024, 0, stream>>>(cntp, candp, pairp, kPairs);
  }

  // 7) write [B, nE, DIM] output
  emit_kernel<<<BATCH * nE, DIM, 0, stream>>>(E, pairp, out, nE, kPairs);
}